// Haar_21612275433974
// MI455X (gfx1250) — compile-verified
//
#include <hip/hip_runtime.h>

// Haar forward DWT: (8,64,512,512) f32 -> (8, 4*64, 256, 256) f32, subband-major.
// Pure streaming kernel (~1.07 GB traffic, ~0.4 GFLOP): optimize the data path,
// not the math. CDNA5 path used: GLOBAL_LOAD_ASYNC_TO_LDS_B128 staging +
// s_wait_asynccnt + ds_load_b128 + non-temporal global_store_b128.

#define BATCH 8
#define IN_C  64
#define H_IN  512
#define W_IN  512
#define H_OUT 256
#define W_OUT 256

typedef float v4f __attribute__((ext_vector_type(4)));
typedef int   v4i __attribute__((ext_vector_type(4)));

#define AS1 __attribute__((address_space(1)))
#define AS3 __attribute__((address_space(3)))

__global__ __launch_bounds__(256) void haar_fwd_kernel(const float* __restrict__ in,
                                                       float* __restrict__ out) {
    // 8 input rows of one (b,c) plane = 8 * 512 * 4B = 16 KB.
    __shared__ float tile[8 * W_IN];

    const int blk  = blockIdx.x;          // 0 .. 32767
    const int ygrp = blk & 63;            // 64 groups of 4 output rows
    const int bc   = blk >> 6;            // b*64 + c, 0..511
    const int c    = bc & 63;
    const int b    = bc >> 6;

    // 8 consecutive input rows of one (b,c) plane are contiguous: 16 KB.
    const float* gsrc = in + (size_t)bc * (H_IN * W_IN) + (size_t)ygrp * (8 * W_IN);

    const int t = threadIdx.x;

#if __has_builtin(__builtin_amdgcn_global_load_async_to_lds_b128)
    // Stage 16 KB = 1024 x 16B chunks; 4 async b128 copies per thread.
    {
        #pragma unroll
        for (int j = 0; j < 4; ++j) {
            const int k = t + j * 256;    // 16-byte chunk index, 0..1023
            __builtin_amdgcn_global_load_async_to_lds_b128(
                (AS1 v4i*)(gsrc + (size_t)k * 4),
                (AS3 v4i*)(&tile[k * 4]),
                0, 0);
        }
    }
  #if __has_builtin(__builtin_amdgcn_s_wait_asynccnt)
    __builtin_amdgcn_s_wait_asynccnt(0);
  #else
    asm volatile("s_wait_asynccnt 0x0" ::: "memory");
  #endif
#else
    // Fallback: direct vector copy through VGPRs.
    {
        const v4f* g4 = (const v4f*)gsrc;
        v4f* l4 = (v4f*)tile;
        #pragma unroll
        for (int j = 0; j < 4; ++j) {
            const int k = t + j * 256;
            l4[k] = g4[k];
        }
    }
#endif
    __syncthreads();

    // Thread layout: r = output row within group (0..3), q = quad of output cols (0..63).
    const int r = t >> 6;
    const int q = t & 63;

    const v4f* row0 = (const v4f*)(tile + (2 * r) * W_IN);
    const v4f* row1 = (const v4f*)(tile + (2 * r + 1) * W_IN);
    const v4f a0 = row0[2 * q];           // top row, input cols 8q .. 8q+3
    const v4f a1 = row0[2 * q + 1];       // top row, input cols 8q+4 .. 8q+7
    const v4f c0 = row1[2 * q];           // bottom row
    const v4f c1 = row1[2 * q + 1];

    v4f ll, lh, hl, hh;
    {   // output col 4q  : inputs (a0.x,a0.y | c0.x,c0.y)
        const float ts = a0.x + a0.y, td = a0.x - a0.y;
        const float us = c0.x + c0.y, ud = c0.x - c0.y;
        ll.x = 0.25f * (ts + us); lh.x = 0.25f * (td + ud);
        hl.x = 0.25f * (ts - us); hh.x = 0.25f * (td - ud);
    }
    {   // output col 4q+1: inputs (a0.z,a0.w | c0.z,c0.w)
        const float ts = a0.z + a0.w, td = a0.z - a0.w;
        const float us = c0.z + c0.w, ud = c0.z - c0.w;
        ll.y = 0.25f * (ts + us); lh.y = 0.25f * (td + ud);
        hl.y = 0.25f * (ts - us); hh.y = 0.25f * (td - ud);
    }
    {   // output col 4q+2: inputs (a1.x,a1.y | c1.x,c1.y)
        const float ts = a1.x + a1.y, td = a1.x - a1.y;
        const float us = c1.x + c1.y, ud = c1.x - c1.y;
        ll.z = 0.25f * (ts + us); lh.z = 0.25f * (td + ud);
        hl.z = 0.25f * (ts - us); hh.z = 0.25f * (td - ud);
    }
    {   // output col 4q+3: inputs (a1.z,a1.w | c1.z,c1.w)
        const float ts = a1.z + a1.w, td = a1.z - a1.w;
        const float us = c1.z + c1.w, ud = c1.z - c1.w;
        ll.w = 0.25f * (ts + us); lh.w = 0.25f * (td + ud);
        hl.w = 0.25f * (ts - us); hh.w = 0.25f * (td - ud);
    }

    // Output layout: ((b*4 + s)*C + c) * 256*256 + y*256 + x  (subband-major).
    const int y = ygrp * 4 + r;
    const size_t plane  = (size_t)H_OUT * W_OUT;
    const size_t sband  = (size_t)IN_C * plane;   // stride between subbands
    float* obase = out + ((size_t)(b * 4) * IN_C + (size_t)c) * plane
                       + (size_t)y * W_OUT + 4 * q;

    // Streaming output: non-temporal b128 stores.
    __builtin_nontemporal_store(ll, (v4f*)(obase));
    __builtin_nontemporal_store(lh, (v4f*)(obase + sband));
    __builtin_nontemporal_store(hl, (v4f*)(obase + 2 * sband));
    __builtin_nontemporal_store(hh, (v4f*)(obase + 3 * sband));
}

extern "C" void kernel_launch(void* const* d_in, const int* in_sizes, int n_in,
                              void* d_out, int out_size, void* d_ws, size_t ws_size,
                              hipStream_t stream) {
    const float* in = (const float*)d_in[0];   // (8, 64, 512, 512) f32
    // d_in[1] = Haar weights: fixed +/-0.25 pattern, hardcoded in the kernel.
    float* out = (float*)d_out;                // (8, 256, 256, 256) f32

    const int grid = BATCH * IN_C * (H_OUT / 4);   // 32768 blocks
    haar_fwd_kernel<<<dim3(grid), dim3(256), 0, stream>>>(in, out);
}